// FastRelPosIds_19361712570452
// MI455X (gfx1250) — compile-verified
//
#include <hip/hip_runtime.h>
#include <stdint.h>

// ---- problem constants (match reference hyperparameters / setup_inputs) ----
#define NUM_HEADS   12
#define REL_MAX     64
#define EMBED       768
#define RADIUS      128
#define HEAD_DIM    (EMBED / NUM_HEADS)   // 64
#define NUM_POS     (2 * REL_MAX + 1)     // 129
#define BL          (RADIUS + 1)          // 129
#define WL          (3 * BL)              // 387
#define BATCH       4
#define SEQ         4096
#define NBQ         ((SEQ + BL - 1) / BL) // 32

#define TILE_ELEMS  (BL * WL)             // 49923 floats
#define TILE_BYTES  (TILE_ELEMS * 4)      // 199692 bytes (fits 320KB WGP LDS)
#define NCOPIES     (BATCH * NUM_HEADS * NBQ) // 1536 identical tiles
#define NWG_A       64
#define COPIES_PER_WG (NCOPIES / NWG_A)   // 24

typedef uint32_t v4u __attribute__((ext_vector_type(4)));
typedef uint32_t v8u __attribute__((ext_vector_type(8)));

// ---------------------------------------------------------------------------
// Kernel A: rel_pos_ids. Build the 129x387 id tile once in LDS, then wave 0
// issues ONE Tensor Data Mover store with descriptor iteration:
//   24 iterations x (195KB LDS tile -> global), global_addr += tile,
//   lds_addr += 0  => pure broadcast replication on the DMA path.
// ---------------------------------------------------------------------------
__global__ void __launch_bounds__(256) relpos_tdm_kernel(float* __restrict__ out0) {
    extern __shared__ float tile[];
    const int wid  = threadIdx.x >> 5;   // 8 waves per block
    const int lane = threadIdx.x & 31;

    // Wave-per-row fill: id = clip((j-BL)-jq, -REL_MAX, REL_MAX) + REL_MAX
    for (int jq = wid; jq < BL; jq += 8) {
        float* rowp = &tile[jq * WL];
        for (int j = lane; j < WL; j += 32) {
            int rel = (j - BL) - jq;
            int id  = rel < -REL_MAX ? -REL_MAX : (rel > REL_MAX ? REL_MAX : rel);
            rowp[j] = (float)(id + REL_MAX);
        }
    }
    __syncthreads();

    if (wid == 0) {
        // LDS byte offset of the tile (low 32 bits of the flat pointer).
        uint32_t lds_addr = (uint32_t)(uintptr_t)(&tile[0]);
        uint64_t gaddr    = (uint64_t)(uintptr_t)out0
                          + (uint64_t)blockIdx.x * (uint64_t)COPIES_PER_WG
                                                 * (uint64_t)TILE_BYTES;

        // ---- D# group 0 (128 bits): count=1, lds_addr, global_addr, type=2 ----
        v4u g0;
        g0[0] = 1u;                                      // count=1 (valid), user mode
        g0[1] = lds_addr;                                 // LDS byte address
        g0[2] = (uint32_t)gaddr;                          // global_addr[31:0]
        g0[3] = ((uint32_t)(gaddr >> 32) & 0x01FFFFFFu)   // global_addr[56:32]
              | 0x80000000u;                              // type = 2 ("image")

        // ---- D# group 1 (256 bits): data_size=4B, 1-D tile, iterate enabled ----
        v8u g1;
        g1[0] = (2u << 16)                               // data_size = 2 (4 bytes)
              | (1u << 19);                              // iterate_enable = 1
        g1[1] = ((uint32_t)(TILE_ELEMS & 0xFFFFu)) << 16; // tensor_dim0[15:0]
        g1[2] = ((uint32_t)(TILE_ELEMS >> 16) & 0xFFFFu)  // tensor_dim0[31:16]
              | (1u << 16);                               // tensor_dim1 = 1
        g1[3] = ((uint32_t)(TILE_ELEMS & 0xFFFFu)) << 16; // tile_dim0 = TILE_ELEMS
        g1[4] = 0u;                                       // tile_dim1/2 unused
        g1[5] = (uint32_t)TILE_ELEMS;                     // tensor_dim0_stride[31:0]
        g1[6] = 0u;                                       // stride hi / dim1_stride lo
        g1[7] = 0u;

        // ---- D# group 2 (128 bits), iterate mode ----
        v4u g2;
        g2[0] = 1u;                                       // tensor_dim2 (unused) = 1
        g2[1] = 0u;                                       // lds_addr_increment = 0
        g2[2] = (uint32_t)TILE_ELEMS;                     // global_addr_increment[31:0]
        g2[3] = 0u                                        // global_addr_increment[47:32]
              | ((uint32_t)(COPIES_PER_WG - 1) << 16);    // iterate_count: 23 -> 24x

        // ---- D# group 3 (128 bits): zeros ----
        v4u g3;
        g3[0] = 0u; g3[1] = 0u; g3[2] = 0u; g3[3] = 0u;

        asm volatile("tensor_store_from_lds %0, %1, %2, %3"
                     :
                     : "s"(g0), "s"(g1), "s"(g2), "s"(g3)
                     : "memory");
        __builtin_amdgcn_s_wait_tensorcnt(0);
    }
}

// ---------------------------------------------------------------------------
// Kernel B: re[bh, p, d] = weight[p, d*NUM_HEADS + (bh % NUM_HEADS)]
// ---------------------------------------------------------------------------
__global__ void __launch_bounds__(256) re_kernel(const float* __restrict__ weight,
                                                 float* __restrict__ out1) {
    const int total = BATCH * NUM_HEADS * NUM_POS * HEAD_DIM;
    int idx = blockIdx.x * 256 + threadIdx.x;
    if (idx >= total) return;
    int d  = idx % HEAD_DIM;
    int p  = (idx / HEAD_DIM) % NUM_POS;
    int bh = idx / (HEAD_DIM * NUM_POS);
    int h  = bh % NUM_HEADS;
    out1[idx] = weight[p * EMBED + d * NUM_HEADS + h];
}

// ---------------------------------------------------------------------------
// Kernel C: mask. One wave32 per (b, q, jq) row of 387; __all() does the
// "fully masked row -> unmask" reduction at wave scope.
// ---------------------------------------------------------------------------
__device__ __forceinline__ bool mask_elem(const unsigned char* kpm,
                                          int b, int q, int jq, int j) {
    int rel   = (j - BL) - jq;
    bool base = (rel > RADIUS) || (rel < -RADIUS);
    int  t    = j + q * BL;                  // index into padded kpm (left pad = BL)
    bool pad;
    if (t < BL || t >= BL + SEQ) pad = true; // left/right padding is True
    else                         pad = (kpm[b * SEQ + (t - BL)] != 0);
    return base || pad;
}

__global__ void __launch_bounds__(256) mask_kernel(const unsigned char* __restrict__ kpm,
                                                   float* __restrict__ out2) {
    const int lane = threadIdx.x & 31;
    const int row  = blockIdx.x * 8 + (threadIdx.x >> 5);
    const int totalRows = BATCH * NBQ * BL;
    if (row >= totalRows) return;

    const int jq = row % BL;
    const int q  = (row / BL) % NBQ;
    const int b  = row / (BL * NBQ);

    // Pass 1: is the entire row masked?
    bool allm = true;
    for (int j = lane; j < WL; j += 32)
        allm = allm && mask_elem(kpm, b, q, jq, j);
    const bool rowAll = (bool)__all(allm);   // wave32-wide AND

    // Pass 2: write (fully-masked rows are forced back to unmasked).
    float* o = out2 + (size_t)row * WL;
    for (int j = lane; j < WL; j += 32)
        o[j] = rowAll ? 0.0f : (mask_elem(kpm, b, q, jq, j) ? 1.0f : 0.0f);
}

// ---------------------------------------------------------------------------
extern "C" void kernel_launch(void* const* d_in, const int* in_sizes, int n_in,
                              void* d_out, int out_size, void* d_ws, size_t ws_size,
                              hipStream_t stream) {
    // setup_inputs order: batch_size, seq_len_q, seq_len_k, key_padding_mask, weight
    const unsigned char* kpm    = (const unsigned char*)d_in[3];
    const float*         weight = (const float*)d_in[4];

    float* out0 = (float*)d_out;                                        // rel_pos_ids
    float* out1 = out0 + (size_t)NCOPIES * TILE_ELEMS;                  // re
    float* out2 = out1 + (size_t)BATCH * NUM_HEADS * NUM_POS * HEAD_DIM;// mask

    // A) rel_pos_ids via TDM iterate-replication: 64 WGs x 1 descriptor x 24 tiles.
    relpos_tdm_kernel<<<NWG_A, 256, TILE_BYTES, stream>>>(out0);

    // B) re gather: 396,288 elements.
    {
        const int total  = BATCH * NUM_HEADS * NUM_POS * HEAD_DIM;
        const int blocks = (total + 255) / 256;
        re_kernel<<<blocks, 256, 0, stream>>>(weight, out1);
    }

    // C) mask: 16,512 rows, 8 wave32 rows per 256-thread block.
    {
        const int rows   = BATCH * NBQ * BL;
        const int blocks = (rows + 7) / 8;
        mask_kernel<<<blocks, 256, 0, stream>>>(kpm, out2);
    }
}